// SSKernelNPLR_63900523430285
// MI455X (gfx1250) — compile-verified
//
#include <hip/hip_runtime.h>

typedef __attribute__((ext_vector_type(2))) float v2f;
typedef __attribute__((ext_vector_type(8))) float v8f;

#define H_DIM   256
#define N_DIM   64
#define L_SEQ   8192
#define LF      4097     // L/2 + 1
#define LF_PAD  4160     // 65 blocks * 4 waves * 16 columns
#define NBLK_X  65       // ceil(257 tiles / 4 waves)

__device__ __forceinline__ float2 cmul(float2 a, float2 b) {
    float2 r; r.x = a.x*b.x - a.y*b.y; r.y = a.x*b.y + a.y*b.x; return r;
}

// ---------------------------------------------------------------------------
// Kernel 1: precompute dt, w*dt, the four v = Cc[a]*Bc[b] products,
// z[l] = 2(1-omega)/(1+omega) and w2[l] = 2/(1+omega).
// ---------------------------------------------------------------------------
__global__ void __launch_bounds__(256)
precompute_kernel(const float* __restrict__ log_dt,
                  const float* __restrict__ C_re, const float* __restrict__ C_im,
                  const float* __restrict__ B_re, const float* __restrict__ B_im,
                  const float* __restrict__ P_re, const float* __restrict__ P_im,
                  const float* __restrict__ Q_re, const float* __restrict__ Q_im,
                  const float* __restrict__ w_re, const float* __restrict__ w_im,
                  float* __restrict__ dtf, float2* __restrict__ zb,
                  float2* __restrict__ w2b, float2* __restrict__ wdtb,
                  float2* __restrict__ vb)
{
    int i = blockIdx.x * blockDim.x + threadIdx.x;

    if (i < H_DIM * N_DIM) {
        int h = i >> 6;
        float dtv = __expf(log_dt[h]);
        float2 wdt; wdt.x = w_re[i] * dtv; wdt.y = w_im[i] * dtv;
        wdtb[i] = wdt;
        float2 B; B.x = B_re[i]; B.y = B_im[i];
        float2 C; C.x = C_re[i]; C.y = C_im[i];
        float2 P; P.x = P_re[i]; P.y = P_im[i];
        float2 Q; Q.x = Q_re[i]; Q.y = Q_im[i];
        vb[i*4 + 0] = cmul(C, B);   // v00
        vb[i*4 + 1] = cmul(C, P);   // v01
        vb[i*4 + 2] = cmul(Q, B);   // v10
        vb[i*4 + 3] = cmul(Q, P);   // v11
    }

    if (i < LF_PAD) {
        float th = 6.283185307179586f * (float)i / (float)L_SEQ;
        float sn, cs; __sincosf(th, &sn, &cs);
        float2 om;  om.x = cs;           om.y = -sn;     // exp(-2*pi*i*l/L)
        float2 opo; opo.x = 1.0f + om.x; opo.y = om.y;   // 1 + omega
        float inv = 1.0f / (opo.x*opo.x + opo.y*opo.y);
        float2 num; num.x = 2.0f * (1.0f - om.x); num.y = -2.0f * om.y;
        float2 z;
        z.x = (num.x*opo.x + num.y*opo.y) * inv;
        z.y = (num.y*opo.x - num.x*opo.y) * inv;
        zb[i] = z;
        float2 w2; w2.x = 2.0f*opo.x*inv; w2.y = -2.0f*opo.y*inv;
        w2b[i] = w2;
    }

    if (i < H_DIM) dtf[i] = __expf(log_dt[i]);
}

// ---------------------------------------------------------------------------
// Kernel 2: Cauchy einsum via realified f32 WMMA (16x16x4), then Woodbury.
// 128 threads = 4 waves per block; block owns one h and 4 adjacent 16-wide
// l-tiles.  A (realified v, 16x128, rows 8-15 zero) is built once in LDS in
// its final row-major layout; each wave builds its own B panel stored
// column-major (Bsh[col][k]) so both fragment fetches are single aligned
// ds_load_b64's and the K-loop is pure ds_load + v_wmma.
// ---------------------------------------------------------------------------
__global__ void __launch_bounds__(128)
cauchy_wmma_kernel(const float*  __restrict__ dtf,
                   const float2* __restrict__ zb,
                   const float2* __restrict__ w2b,
                   const float2* __restrict__ wdtb,
                   const float2* __restrict__ vb,
                   float2* __restrict__ kfb)
{
    __shared__ float Ash[16 * 128];        // realified A, 8 KB
    __shared__ float Bsh[4][16 * 128];     // per-wave realified B^T, 32 KB

    const int tid  = threadIdx.x;
    const int lane = tid & 31;
    const int w    = tid >> 5;             // wave id 0..3
    const int h    = blockIdx.y;
    const int l0   = (blockIdx.x * 4 + w) * 16;

    __builtin_prefetch(vb + h * N_DIM * 4, 0, 0);   // global_prefetch_b8

    // ---- build realified A in final layout (select logic runs once) ----
    for (int idx = tid; idx < 16 * 128; idx += 128) {
        int   mm = idx >> 7;               // row 0..15
        int   k  = idx & 127;
        float val = 0.0f;
        if (mm < 8) {
            int  rr = mm & 3;              // which ab product
            bool im = (mm & 4) != 0;       // rows 4-7 carry Im(v)
            int  n  = k & 63;
            bool hi = (k >= 64);           // columns 64-127 = Im(recip) block
            float2 v = vb[h * N_DIM * 4 + n * 4 + rr];
            val = im ? (hi ? v.x : v.y) : (hi ? -v.y : v.x);
        }
        Ash[idx] = val;
    }

    // ---- reciprocals -> B^T panel: Bsh[j][n] = Re, Bsh[j][64+n] = Im ----
    for (int idx = lane; idx < N_DIM * 16; idx += 32) {
        int n = idx >> 4, j = idx & 15;
        float2 z  = zb[l0 + j];
        float2 wv = wdtb[h * N_DIM + n];
        float dx = z.x - wv.x, dy = z.y - wv.y;
        float inv = 1.0f / (dx*dx + dy*dy);
        Bsh[w][j * 128 + n]      =  dx * inv;
        Bsh[w][j * 128 + 64 + n] = -dy * inv;
    }
    __syncthreads();

    // ---- K loop: 32 x (2 ds_load_b64 + v_wmma_f32_16x16x4_f32) ----
    const int m  = lane & 15;
    const int hf = lane >> 4;
    const float* Arow = &Ash[m * 128];
    const float* Brow = &Bsh[w][(lane & 15) * 128];

    v8f acc = {0.f, 0.f, 0.f, 0.f, 0.f, 0.f, 0.f, 0.f};
    #pragma unroll
    for (int kk = 0; kk < 32; ++kk) {
        const int k0 = kk * 4 + hf * 2;        // even -> 8B aligned
        float2 af = *(const float2*)&Arow[k0];
        float2 bf = *(const float2*)&Brow[k0];
        v2f a; a.x = af.x; a.y = af.y;
        v2f b; b.x = bf.x; b.y = bf.y;
        acc = __builtin_amdgcn_wmma_f32_16x16x4_f32(false, a, false, b,
                                                    (short)0, acc, false, false);
    }

    // ---- epilogue: lanes 0-15 own one l column each ----
    if (lane < 16) {
        int l = l0 + lane;
        if (l < LF) {
            float dtv = dtf[h];
            float2 r00, r01, r10, r11;
            r00.x = acc[0]*dtv; r00.y = acc[4]*dtv;
            r01.x = acc[1]*dtv; r01.y = acc[5]*dtv;
            r10.x = acc[2]*dtv; r10.y = acc[6]*dtv;
            r11.x = acc[3]*dtv; r11.y = acc[7]*dtv;

            float2 den; den.x = 1.0f + r11.x; den.y = r11.y;
            float2 num = cmul(r01, r10);
            float dinv = 1.0f / (den.x*den.x + den.y*den.y);
            float2 q;
            q.x = (num.x*den.x + num.y*den.y) * dinv;
            q.y = (num.y*den.x - num.x*den.y) * dinv;

            float2 kf; kf.x = r00.x - q.x; kf.y = r00.y - q.y;
            kf = cmul(kf, w2b[l]);          // * 2/(1+omega)
            kfb[h * LF + l] = kf;
        }
    }
}

// ---------------------------------------------------------------------------
// Kernel 3: irfft(k_f, n=8192) per h.
// Real-packing: Y[k] = (K[k]+conj(K[L2-k])) + i*e^{2pi i k/L}*(K[k]-conj(K[L2-k]))
// then 4096-pt inverse complex DFT (unscaled) via 12-stage Stockham radix-2
// in LDS; x[2m]=Re/L, x[2m+1]=Im/L.
// ---------------------------------------------------------------------------
__global__ void __launch_bounds__(256)
irfft_kernel(const float2* __restrict__ kfb, float* __restrict__ out)
{
    __shared__ float2 bufA[4096];   // 32 KB
    __shared__ float2 bufB[4096];   // 32 KB
    const int tid = threadIdx.x;
    const int h   = blockIdx.x;
    const float TWO_PI = 6.283185307179586f;

    __builtin_prefetch(kfb + h * LF + tid, 0, 0);

    // Build packed spectrum Y into bufA
    for (int k = tid; k < 4096; k += 256) {
        float2 Ka = kfb[h * LF + k];
        float2 Kb = kfb[h * LF + (4096 - k)];
        float2 s; s.x = Ka.x + Kb.x; s.y = Ka.y - Kb.y;   // K + conj(K')
        float2 d; d.x = Ka.x - Kb.x; d.y = Ka.y + Kb.y;   // K - conj(K')
        float th = TWO_PI * (float)k / (float)L_SEQ;
        float sn, cs; __sincosf(th, &sn, &cs);
        float2 y;
        y.x = s.x - (cs*d.y + sn*d.x);                    // s + i*w*d
        y.y = s.y + (cs*d.x - sn*d.y);
        bufA[k] = y;
    }
    __syncthreads();

    // Stockham DIF, inverse twiddles (+sin), ping-pong LDS buffers.
    float2* cur = bufA;
    float2* nxt = bufB;
    int nn = 4096;
    for (int stage = 0; stage < 12; ++stage) {
        const int m   = nn >> 1;
        const int str = 1 << stage;
        for (int i = tid; i < 2048; i += 256) {
            int p = i >> stage;
            int q = i & (str - 1);
            float th = TWO_PI * (float)p / (float)nn;
            float sn, cs; __sincosf(th, &sn, &cs);
            float2 a = cur[q + str * p];
            float2 b = cur[q + str * (p + m)];
            float2 apb; apb.x = a.x + b.x; apb.y = a.y + b.y;
            float2 amb; amb.x = a.x - b.x; amb.y = a.y - b.y;
            float2 t; t.x = amb.x*cs - amb.y*sn; t.y = amb.x*sn + amb.y*cs;
            nxt[q + str * (2*p)]     = apb;
            nxt[q + str * (2*p + 1)] = t;
        }
        __syncthreads();
        float2* tmp = cur; cur = nxt; nxt = tmp;
        nn = m;
    }

    const float invL = 1.0f / (float)L_SEQ;
    for (int i = tid; i < 4096; i += 256) {
        float2 y = cur[i];
        out[h * L_SEQ + 2*i]     = y.x * invL;
        out[h * L_SEQ + 2*i + 1] = y.y * invL;
    }
}

// ---------------------------------------------------------------------------
extern "C" void kernel_launch(void* const* d_in, const int* in_sizes, int n_in,
                              void* d_out, int out_size, void* d_ws, size_t ws_size,
                              hipStream_t stream)
{
    (void)in_sizes; (void)n_in; (void)out_size; (void)ws_size;

    const float* log_dt = (const float*)d_in[0];
    const float* C_re   = (const float*)d_in[1];
    const float* C_im   = (const float*)d_in[2];
    const float* B_re   = (const float*)d_in[3];
    const float* B_im   = (const float*)d_in[4];
    const float* P_re   = (const float*)d_in[5];
    const float* P_im   = (const float*)d_in[6];
    const float* Q_re   = (const float*)d_in[7];
    const float* Q_im   = (const float*)d_in[8];
    const float* w_re   = (const float*)d_in[9];
    const float* w_im   = (const float*)d_in[10];
    // d_in[11] = L (int, known 8192 at compile time)

    // Workspace carve-up (floats):
    //   dtf   : 256
    //   zb    : LF_PAD float2
    //   w2b   : LF_PAD float2
    //   wdtb  : H*N float2
    //   vb    : H*N*4 float2
    //   kfb   : H*LF float2   (~8.4 MB)
    float*  ws   = (float*)d_ws;
    float*  dtf  = ws;
    float2* zb   = (float2*)(ws + 256);
    float2* w2b  = zb + LF_PAD;
    float2* wdtb = w2b + LF_PAD;
    float2* vb   = wdtb + (H_DIM * N_DIM);
    float2* kfb  = vb + (H_DIM * N_DIM * 4);

    precompute_kernel<<<64, 256, 0, stream>>>(log_dt, C_re, C_im, B_re, B_im,
                                              P_re, P_im, Q_re, Q_im, w_re, w_im,
                                              dtf, zb, w2b, wdtb, vb);

    cauchy_wmma_kernel<<<dim3(NBLK_X, H_DIM), 128, 0, stream>>>(dtf, zb, w2b,
                                                                wdtb, vb, kfb);

    irfft_kernel<<<H_DIM, 256, 0, stream>>>(kfb, (float*)d_out);
}